// OrdinalLabelSmoothingLoss_9861244911665
// MI455X (gfx1250) — compile-verified
//
#include <hip/hip_runtime.h>
#include <cstdint>

// ---------------------------------------------------------------------------
// OrdinalLabelSmoothingLoss for MI455X (gfx1250, wave32).
//
// loss_row = W*lse - ( far*sum(x) + sum_{|d|<=2,valid}(w_d-far)*x[t+d] )
//   lse = rowmax + log(sum exp(x - rowmax)),  W = far*C + sum_valid(w_d-far)
// out = mean(loss_row)
//
// Bandwidth-bound (128 MB @ 23.3 TB/s ~ 5.5us). Data path: async global->LDS
// (ASYNCcnt) staging of each 16KB row, then two LDS passes (max+sum, sumexp).
// ---------------------------------------------------------------------------

#define NT 256
constexpr int C = 4096;

#if defined(__AMDGCN__) && \
    __has_builtin(__builtin_amdgcn_global_load_async_to_lds_b128) && \
    __has_builtin(__builtin_amdgcn_s_wait_asynccnt)
#define USE_ASYNC_LDS 1
#else
#define USE_ASYNC_LDS 0
#endif

#define AS1 __attribute__((address_space(1)))
#define AS3 __attribute__((address_space(3)))
typedef int v4i __attribute__((vector_size(4 * sizeof(int))));

// 16-byte global -> LDS copy. On gfx1250 this is GLOBAL_LOAD_ASYNC_TO_LDS_B128
// (tracked by ASYNCcnt; data never touches VGPRs). Fallback: plain copy.
__device__ __forceinline__ void g2l_b128(const float* g, float* l) {
#if USE_ASYNC_LDS
  __builtin_amdgcn_global_load_async_to_lds_b128((AS1 v4i*)g, (AS3 v4i*)l,
                                                 /*offset=*/0, /*cpol=*/0);
#else
  *(float4*)l = *(const float4*)g;
#endif
}

__device__ __forceinline__ void wait_async_all() {
#if USE_ASYNC_LDS
  __builtin_amdgcn_s_wait_asynccnt(0);
#endif
}

__global__ void __launch_bounds__(NT)
ordinal_ls_row_kernel(const float* __restrict__ pred,
                      const long long* __restrict__ targ,
                      float* __restrict__ row_loss, int rows) {
  __shared__ __align__(16) float tile[C];        // one full row: 16 KB
  __shared__ float sm[NT / 32], ssx[NT / 32], sse[NT / 32];

  const int row = blockIdx.x;
  if (row >= rows) return;
  const int tid = threadIdx.x;
  const int lane = tid & 31;   // wave32
  const int wid = tid >> 5;
  const float* rowp = pred + (size_t)row * C;

  // ---- stage row into LDS (async on gfx1250, 128-bit fully coalesced) ----
  constexpr int ITERS = C / (NT * 4);            // 4 x float4 per thread
#pragma unroll
  for (int i = 0; i < ITERS; ++i) {
    const int off = (i * NT + tid) * 4;
    g2l_b128(rowp + off, &tile[off]);
  }
  wait_async_all();      // per-wave: my async writes to LDS landed
  __syncthreads();       // cross-wave visibility of the staged row

  // ---- pass 1: row max and plain row sum ----
  float m = -3.402823466e+38f;
  float sx = 0.f;
#pragma unroll
  for (int i = 0; i < ITERS; ++i) {
    const float4 v = *(const float4*)&tile[(i * NT + tid) * 4];
    m = fmaxf(m, fmaxf(fmaxf(v.x, v.y), fmaxf(v.z, v.w)));
    sx += (v.x + v.y) + (v.z + v.w);
  }
#pragma unroll
  for (int o = 16; o > 0; o >>= 1) {             // wave32 butterfly
    m = fmaxf(m, __shfl_xor(m, o, 32));
    sx += __shfl_xor(sx, o, 32);
  }
  if (lane == 0) { sm[wid] = m; ssx[wid] = sx; }
  __syncthreads();
  float rmax = sm[0], rsx = ssx[0];
#pragma unroll
  for (int w = 1; w < NT / 32; ++w) { rmax = fmaxf(rmax, sm[w]); rsx += ssx[w]; }

  // ---- pass 2: sum exp(x - rowmax) (one exp per element, LDS resident) ----
  float se = 0.f;
#pragma unroll
  for (int i = 0; i < ITERS; ++i) {
    const float4 v = *(const float4*)&tile[(i * NT + tid) * 4];
    se += (__expf(v.x - rmax) + __expf(v.y - rmax)) +
          (__expf(v.z - rmax) + __expf(v.w - rmax));
  }
#pragma unroll
  for (int o = 16; o > 0; o >>= 1) se += __shfl_xor(se, o, 32);
  if (lane == 0) sse[wid] = se;
  __syncthreads();

  // ---- combine: 5 target-neighborhood taps come straight from LDS ----
  if (tid == 0) {
    float sumexp = 0.f;
#pragma unroll
    for (int w = 0; w < NT / 32; ++w) sumexp += sse[w];
    const float lse = rmax + logf(sumexp);

    const long long t = targ[row];
    const float far = (float)(0.1 * 0.1 / (double)(C - 3));
    const float wtab[3] = {(float)(1.0 - 0.1),   // dist 0
                           (float)(0.1 * 0.4),   // dist 1
                           (float)(0.1 * 0.2)};  // dist 2
    float W = far * (float)C;
    float dot = far * rsx;
#pragma unroll
    for (int d = -2; d <= 2; ++d) {
      const long long c = t + d;
      if (c >= 0 && c < C) {
        const float dw = wtab[d < 0 ? -d : d] - far;
        dot += dw * tile[(int)c];
        W += dw;
      }
    }
    row_loss[row] = W * lse - dot;               // = -(sum w*logp)
  }
}

// Deterministic fixed-order mean reduction (no float atomics -> replay-stable).
__global__ void __launch_bounds__(NT)
mean_kernel(const float* __restrict__ v, float* __restrict__ out, int n) {
  __shared__ float s[NT];
  float acc = 0.f;
  for (int i = threadIdx.x; i < n; i += NT) acc += v[i];
  s[threadIdx.x] = acc;
  __syncthreads();
  for (int o = NT / 2; o > 0; o >>= 1) {
    if ((int)threadIdx.x < o) s[threadIdx.x] += s[threadIdx.x + o];
    __syncthreads();
  }
  if (threadIdx.x == 0) out[0] = s[0] / (float)n;
}

extern "C" void kernel_launch(void* const* d_in, const int* in_sizes, int n_in,
                              void* d_out, int out_size, void* d_ws,
                              size_t ws_size, hipStream_t stream) {
  (void)n_in; (void)out_size; (void)ws_size;
  const float* pred = (const float*)d_in[0];
  const long long* targ = (const long long*)d_in[1];   // int64 targets
  float* out = (float*)d_out;
  float* ws = (float*)d_ws;                            // rows floats of scratch
  const int rows = in_sizes[1];

  ordinal_ls_row_kernel<<<rows, NT, 0, stream>>>(pred, targ, ws, rows);
  mean_kernel<<<1, NT, 0, stream>>>(ws, out, rows);
}